// GNN_node_79894981640400
// MI455X (gfx1250) — compile-verified
//
#include <hip/hip_runtime.h>
#include <hip/hip_bf16.h>

typedef __attribute__((ext_vector_type(2))) float v2f;
typedef __attribute__((ext_vector_type(8))) float v8f;

#define N_NODES 50000
#define N_EDGES 640000
#define DIM     128
#define DIM2    256
#define NLAYER  3

__device__ __forceinline__ void atomicAddF32(float* p, float v) {
  // relaxed agent-scope fp32 add -> global_atomic_add_f32 (no return)
  __hip_atomic_fetch_add(p, v, __ATOMIC_RELAXED, __HIP_MEMORY_SCOPE_AGENT);
}

// ---------------------------------------------------------------------------
// Edge encode + gather + ReLU + scatter-add.  One wave32 per edge, 4 channels
// per lane.  We (8x128) and be cached in LDS.
// ---------------------------------------------------------------------------
__global__ __launch_bounds__(256)
void gin_edge_scatter(const float* __restrict__ h,
                      const int* __restrict__ srcIdx,
                      const int* __restrict__ dstIdx,
                      const float* __restrict__ edge_attr,
                      const float* __restrict__ We,
                      const float* __restrict__ be,
                      float* __restrict__ agg) {
  __shared__ float sW[8 * DIM];
  __shared__ float sb[DIM];
  for (int i = threadIdx.x; i < 8 * DIM; i += 256) sW[i] = We[i];
  if (threadIdx.x < DIM) sb[threadIdx.x] = be[threadIdx.x];
  __syncthreads();

  const int lane = threadIdx.x & 31;
  const int c    = lane * 4;
  const int e    = blockIdx.x * 8 + (threadIdx.x >> 5);
  if (e >= N_EDGES) return;

  float ea[8];
#pragma unroll
  for (int k = 0; k < 8; ++k) ea[k] = edge_attr[(size_t)e * 8 + k];
  const int s = srcIdx[e];
  const int d = dstIdx[e];

  const float4 hv = *reinterpret_cast<const float4*>(h + (size_t)s * DIM + c);
  float hvals[4] = {hv.x, hv.y, hv.z, hv.w};

#pragma unroll
  for (int j = 0; j < 4; ++j) {
    float acc = sb[c + j];
#pragma unroll
    for (int k = 0; k < 8; ++k) acc = fmaf(ea[k], sW[k * DIM + c + j], acc);
    float m = hvals[j] + acc;
    m = m > 0.0f ? m : 0.0f;
    atomicAddF32(&agg[(size_t)d * DIM + c + j], m);
  }
}

// ---------------------------------------------------------------------------
// z = (1+eps)*h + agg   (in place into agg), float4 vectorized
// ---------------------------------------------------------------------------
__global__ __launch_bounds__(256)
void gin_combine(const float* __restrict__ h, float* __restrict__ z,
                 const float* __restrict__ eps_l, int nvec4) {
  const int i = blockIdx.x * 256 + threadIdx.x;
  if (i >= nvec4) return;
  const float s = 1.0f + *eps_l;
  const float4 hv = reinterpret_cast<const float4*>(h)[i];
  float4 zv = reinterpret_cast<float4*>(z)[i];
  zv.x = fmaf(s, hv.x, zv.x);
  zv.y = fmaf(s, hv.y, zv.y);
  zv.z = fmaf(s, hv.z, zv.z);
  zv.w = fmaf(s, hv.w, zv.w);
  reinterpret_cast<float4*>(z)[i] = zv;
}

// ---------------------------------------------------------------------------
// Y[N x ncols] = A[N x K] * W[K x ncols] + bias, fp32 WMMA 16x16x4.
// Block = 256 threads = 8 waves.  Wave w owns TPW 16x16 tiles at columns
// c = w*16 + t*128, so one block covers 16 rows x (TPW*128) cols and one
// A-fragment feeds TPW WMMAs.  The 16-row A tile spans full K (contiguous
// 16*K chunk) and is staged into LDS with CDNA5 async global->LDS B128
// copies (ASYNCcnt), then s_wait_asynccnt + barrier.
// A frag (16x4 f32): lanes 0-15 hold K={k0,k0+1}, lanes 16-31 K={k0+2,k0+3}.
// C/D: lane half selects M base (0 or 8), VGPR r -> M = r + 8*half.
// ---------------------------------------------------------------------------
template <int K, int TPW>
__global__ __launch_bounds__(256)
void gemm_wmma_f32(const float* __restrict__ A, const float* __restrict__ W,
                   const float* __restrict__ bias, float* __restrict__ Y,
                   int ncols) {
  __shared__ float lds[16 * K];
  const int tid = threadIdx.x;
  const int r0  = blockIdx.x * 16;

  // ---- async global -> LDS staging of the 16xK A tile (B128 per lane) ----
  {
    const float* gsrc = A + (size_t)r0 * K;
    constexpr int NV = 16 * K / 4;  // float4 count (512 or 1024)
#pragma unroll
    for (int j = 0; j < NV / 256; ++j) {
      const int i = tid + j * 256;
      const unsigned lds_off = (unsigned)(size_t)(&lds[i * 4]);
      const float* ga = gsrc + i * 4;
      asm volatile("global_load_async_to_lds_b128 %0, %1, off"
                   :
                   : "v"(lds_off), "v"(ga)
                   : "memory");
    }
    asm volatile("s_wait_asynccnt 0x0" ::: "memory");
  }
  __syncthreads();

  const int wave = tid >> 5;
  const int lane = tid & 31;
  const int n    = lane & 15;   // col within 16-wide tile / row M for A frag
  const int kh   = lane >> 4;   // lane-half
  const int cw   = wave * 16;

  v8f acc[TPW];
#pragma unroll
  for (int t = 0; t < TPW; ++t) acc[t] = {};

  const float* lrow = lds + n * K;
  for (int k0 = 0; k0 < K; k0 += 4) {
    v2f a;
    a.x = lrow[k0 + 2 * kh];
    a.y = lrow[k0 + 2 * kh + 1];
    const float* wrow = W + (size_t)(k0 + 2 * kh) * ncols + n;
#pragma unroll
    for (int t = 0; t < TPW; ++t) {
      v2f b;
      const float* wp = wrow + cw + t * 128;
      b.x = wp[0];
      b.y = wp[ncols];
      acc[t] = __builtin_amdgcn_wmma_f32_16x16x4_f32(false, a, false, b,
                                                     (short)0, acc[t], false,
                                                     false);
    }
  }

#pragma unroll
  for (int t = 0; t < TPW; ++t) {
    const int c = cw + t * 128 + n;
    const float bc = bias[c];
#pragma unroll
    for (int r = 0; r < 8; ++r) {
      Y[(size_t)(r0 + r + 8 * kh) * ncols + c] = acc[t][r] + bc;
    }
  }
}

// ---------------------------------------------------------------------------
// BatchNorm column statistics: thread = column (coalesced), rows partitioned
// across blocks, partial sums accumulated with f32 atomics.
// ---------------------------------------------------------------------------
__global__ void col_stats_partial(const float* __restrict__ X,
                                  float* __restrict__ sums,
                                  float* __restrict__ sumsq, int nrows) {
  const int C = blockDim.x;
  const int c = threadIdx.x;
  const int rows_per = (nrows + gridDim.x - 1) / gridDim.x;
  const int rb = blockIdx.x * rows_per;
  const int re = rb + rows_per < nrows ? rb + rows_per : nrows;
  float s = 0.0f, s2 = 0.0f;
  for (int r = rb; r < re; ++r) {
    const float v = X[(size_t)r * C + c];
    s += v;
    s2 = fmaf(v, v, s2);
  }
  atomicAddF32(&sums[c], s);
  atomicAddF32(&sumsq[c], s2);
}

__global__ void col_stats_finalize(const float* __restrict__ sums,
                                   const float* __restrict__ sumsq,
                                   float* __restrict__ mu,
                                   float* __restrict__ rstd, float invN) {
  const int c = threadIdx.x;
  const float m = sums[c] * invN;
  const float v = sumsq[c] * invN - m * m;
  mu[c]   = m;
  rstd[c] = rsqrtf(v + 1e-5f);
}

// ---------------------------------------------------------------------------
// y = (x - mu)*rstd*g + b, optional ReLU.  cmask = C-1 (C is a power of 2).
// ---------------------------------------------------------------------------
__global__ __launch_bounds__(256)
void bn_act(const float* __restrict__ X, float* __restrict__ Y,
            const float* __restrict__ mu, const float* __restrict__ rstd,
            const float* __restrict__ g, const float* __restrict__ b,
            int cmask, size_t total, int relu) {
  const size_t i = (size_t)blockIdx.x * 256 + threadIdx.x;
  if (i >= total) return;
  const int c = (int)(i & (size_t)cmask);
  float y = fmaf((X[i] - mu[c]) * rstd[c], g[c], b[c]);
  if (relu) y = fmaxf(y, 0.0f);
  Y[i] = y;
}

// ---------------------------------------------------------------------------
extern "C" void kernel_launch(void* const* d_in, const int* in_sizes, int n_in,
                              void* d_out, int out_size, void* d_ws,
                              size_t ws_size, hipStream_t stream) {
  const float* x   = (const float*)d_in[0];
  const int*   ei  = (const int*)d_in[1];
  const float* eatt= (const float*)d_in[2];
  const float* eps = (const float*)d_in[3];
  const float* We  = (const float*)d_in[4];
  const float* be  = (const float*)d_in[5];
  const float* W1  = (const float*)d_in[6];
  const float* b1  = (const float*)d_in[7];
  const float* g1  = (const float*)d_in[8];
  const float* bt1 = (const float*)d_in[9];
  const float* W2  = (const float*)d_in[10];
  const float* b2  = (const float*)d_in[11];
  const float* gbn = (const float*)d_in[12];
  const float* bbn = (const float*)d_in[13];
  float* out = (float*)d_out;

  const int* srcIdx = ei;
  const int* dstIdx = ei + N_EDGES;

  char* ws = (char*)d_ws;
  const size_t ND  = (size_t)N_NODES * DIM;   // 6.4M
  const size_t ND2 = (size_t)N_NODES * DIM2;  // 12.8M
  float* agg  = (float*)(ws);                          // N*D   (also z, y2)
  float* y1   = (float*)(ws + ND * 4);                 // N*2D
  float* hbuf = (float*)(ws + ND * 4 + ND2 * 4);       // N*D
  float* sums = (float*)(ws + ND * 4 + ND2 * 4 + ND * 4);
  float* sumsq = sums + DIM2;
  float* mu    = sums + 2 * DIM2;
  float* rstd  = sums + 3 * DIM2;

  const float* hcur = x;
  for (int l = 0; l < NLAYER; ++l) {
    // h buffers ping-pong through d_out: l0 -> out, l1 -> hbuf, l2 -> out
    float* hnext = (l == 1) ? hbuf : out;

    // 1) agg = segment_sum(relu(h[src] + edge_attr@We + be), dst)
    hipMemsetAsync(agg, 0, ND * 4, stream);
    gin_edge_scatter<<<N_EDGES / 8, 256, 0, stream>>>(
        hcur, srcIdx, dstIdx, eatt, We + (size_t)l * 8 * DIM,
        be + (size_t)l * DIM, agg);

    // 2) z = (1+eps)*h + agg   (in place into agg)
    gin_combine<<<(int)((ND / 4 + 255) / 256), 256, 0, stream>>>(
        hcur, agg, eps + l, (int)(ND / 4));

    // 3) y1 = z @ W1 + b1   [N, 256]   (TPW=2 -> one block = all 256 cols)
    gemm_wmma_f32<DIM, 2><<<dim3(N_NODES / 16, 1), 256, 0, stream>>>(
        agg, W1 + (size_t)l * DIM * DIM2, b1 + (size_t)l * DIM2, y1, DIM2);

    // 4) BN(y1) + ReLU, in place
    hipMemsetAsync(sums, 0, 2 * DIM2 * 4, stream);
    col_stats_partial<<<128, DIM2, 0, stream>>>(y1, sums, sumsq, N_NODES);
    col_stats_finalize<<<1, DIM2, 0, stream>>>(sums, sumsq, mu, rstd,
                                               1.0f / N_NODES);
    bn_act<<<(int)((ND2 + 255) / 256), 256, 0, stream>>>(
        y1, y1, mu, rstd, g1 + (size_t)l * DIM2, bt1 + (size_t)l * DIM2,
        DIM2 - 1, ND2, 1);

    // 5) y2 = y1 @ W2 + b2   [N, 128]  (reuse agg buffer, TPW=1)
    gemm_wmma_f32<DIM2, 1><<<dim3(N_NODES / 16, 1), 256, 0, stream>>>(
        y1, W2 + (size_t)l * DIM2 * DIM, b2 + (size_t)l * DIM, agg, DIM);

    // 6) outer BN (+ReLU except last layer) -> hnext
    hipMemsetAsync(sums, 0, 2 * DIM2 * 4, stream);
    col_stats_partial<<<128, DIM, 0, stream>>>(agg, sums, sumsq, N_NODES);
    col_stats_finalize<<<1, DIM, 0, stream>>>(sums, sumsq, mu, rstd,
                                              1.0f / N_NODES);
    bn_act<<<(int)((ND + 255) / 256), 256, 0, stream>>>(
        agg, hnext, mu, rstd, gbn + (size_t)l * DIM, bbn + (size_t)l * DIM,
        DIM - 1, ND, (l < NLAYER - 1) ? 1 : 0);

    hcur = hnext;
  }
}